// GraphAttentionLayer_79474074845764
// MI455X (gfx1250) — compile-verified
//
#include <hip/hip_runtime.h>

typedef float v2f __attribute__((ext_vector_type(2)));
typedef float v8f __attribute__((ext_vector_type(8)));

#define IN_F     256
#define OUT_F    256
#define NNODES   2048
#define NBATCH   8
#define ALPHA    0.01f
#define NEG_INF  -1000000000.0f

#define NJ_CHUNK 512
#define P_STRIDE 516   // 512 + 4 dword pad; stride % 64 banks == 4 -> conflict-free b64 frag reads

// ---------------------------------------------------------------------------
// Kernel 1: h = inp @ W
// Block = 512 threads = 16 waves; block covers 16 rows x 256 cols, wave w owns
// cols [16w,16w+16). 64 x wmma_f32_16x16x4 per wave.
//
// A frag (16x4 f32): lane L -> M = L%16; VGPR0 = A[M][k+(L<16?0:2)], VGPR1 = +1
// B frag (4x16 f32): VGPR0: lanes0-15 K=k, lanes16-31 K=k+2; N = L%16
// C/D (16x16 f32): VGPR r: lanes0-15 M=r, lanes16-31 M=8+r; N = L%16
// ---------------------------------------------------------------------------
__global__ __launch_bounds__(512)
void gat_h_kernel(const float* __restrict__ inp, const float* __restrict__ W,
                  float* __restrict__ h) {
    const int tid  = threadIdx.x;
    const int wave = tid >> 5;
    const int lane = tid & 31;
    const int col  = lane & 15;
    const int hi   = lane >> 4;
    const int koff = hi * 2;

    const int b  = blockIdx.x >> 7;
    const int mt = blockIdx.x & 127;
    const int m0 = mt * 16;
    const int n0 = wave * 16;

    const float* Arow = inp + ((size_t)b * NNODES + (size_t)(m0 + col)) * IN_F;
    const float* Bcol = W + (n0 + col);

    v8f acc = {};
    for (int k = 0; k < IN_F; k += 4) {
        v2f a;
        a.x = Arow[k + koff];
        a.y = Arow[k + koff + 1];
        v2f bb;
        bb.x = Bcol[(size_t)(k + koff)     * OUT_F];
        bb.y = Bcol[(size_t)(k + koff + 1) * OUT_F];
        acc = __builtin_amdgcn_wmma_f32_16x16x4_f32(false, a, false, bb,
                                                    (short)0, acc, false, false);
    }

    float* hout = h + (size_t)b * NNODES * OUT_F;
    for (int r = 0; r < 8; ++r) {
        const int row = m0 + r + hi * 8;
        hout[(size_t)row * OUT_F + n0 + col] = acc[r];
    }
}

// ---------------------------------------------------------------------------
// Kernel 2: e_src[b,n] = h[b,n,:] . a[:256], e_dst[b,n] = h[b,n,:] . a[256:]
// One wave32 per (b,n) row; shfl_xor tree reduction.
// ---------------------------------------------------------------------------
__global__ __launch_bounds__(256)
void gat_e_kernel(const float* __restrict__ h, const float* __restrict__ avec,
                  float* __restrict__ esrc, float* __restrict__ edst) {
    const int wave = blockIdx.x * (blockDim.x >> 5) + (threadIdx.x >> 5);
    const int lane = threadIdx.x & 31;
    if (wave >= NBATCH * NNODES) return;

    const float* hrow = h + (size_t)wave * OUT_F;
    float s1 = 0.f, s2 = 0.f;
    for (int f = lane; f < OUT_F; f += 32) {
        const float hv = hrow[f];
        s1 += hv * avec[f];
        s2 += hv * avec[OUT_F + f];
    }
    for (int off = 16; off > 0; off >>= 1) {
        s1 += __shfl_xor(s1, off, 32);
        s2 += __shfl_xor(s2, off, 32);
    }
    if (lane == 0) { esrc[wave] = s1; edst[wave] = s2; }
}

// ---------------------------------------------------------------------------
// Kernel 3: fused masked softmax + (att @ h) + ReLU, LDS producer/consumer.
// Block = 512 threads = 16 waves <-> (batch b, 16-row i-tile).
//
// Phase 1: wave w computes row-max m_w for row i0+w (shfl_xor reduce -> LDS).
// Phase 2: 4 chunks of 512 j. Per chunk: wave w writes unnormalized
//   P[w][j] = exp(v - m_w) into LDS ONCE (accumulating its row denominator as
//   a per-lane partial -> no separate sum-sweep), barrier, then every wave's
//   hot loop is ds_load_b64 (A frag) + 2 global b32 (B frag, disjoint
//   f-columns per wave) + v_wmma_f32_16x16x4. 512 WMMAs/wave total.
// Normalize by d and ReLU on store. Fully-masked rows degenerate to
// m = -1e9 => exp(0) = 1 everywhere => uniform attention, matching the
// reference's softmax over all-(-1e9) logits.
// ---------------------------------------------------------------------------
__global__ __launch_bounds__(512)
void gat_attn_kernel(const float* __restrict__ h, const int* __restrict__ adj,
                     const float* __restrict__ esrc, const float* __restrict__ edst,
                     float* __restrict__ out) {
    __shared__ __align__(16) float Pl[16 * P_STRIDE];   // 33 KB
    __shared__ float lds_m[16];
    __shared__ float lds_d[16];

    const int tid  = threadIdx.x;
    const int wave = tid >> 5;
    const int lane = tid & 31;
    const int b    = blockIdx.x >> 7;
    const int it   = blockIdx.x & 127;
    const int i0   = it * 16;

    const float* Esrc = esrc + b * NNODES;
    const float* Edst = edst + b * NNODES;

    // Producer role: wave w owns row i0 + w.
    const int    prow  = i0 + wave;
    const float  psv   = Esrc[prow];
    const int*   parow = adj + (size_t)prow * NNODES;

    // ---- Phase 1: row max ----
    {
        float mx = -3.402823466e38f;
        for (int j = lane; j < NNODES; j += 32) {
            const float s  = psv + Edst[j];
            const float lr = s > 0.f ? s : ALPHA * s;
            const float v  = (parow[j] != 0) ? lr : NEG_INF;
            mx = fmaxf(mx, v);
        }
        for (int off = 16; off > 0; off >>= 1)
            mx = fmaxf(mx, __shfl_xor(mx, off, 32));
        if (lane == 0) lds_m[wave] = mx;
    }
    __syncthreads();
    const float pmx = lds_m[wave];

    // Consumer role: wave w owns output f-columns [16w, 16w+16).
    const int col  = lane & 15;
    const int hi   = lane >> 4;
    const int koff = hi * 2;
    const int n0   = wave * 16;
    const float* hb = h + (size_t)b * NNODES * OUT_F + n0 + col;

    v8f   acc  = {};
    float dsum = 0.f;

    for (int jc = 0; jc < NNODES; jc += NJ_CHUNK) {
        __syncthreads();            // previous chunk fully consumed by all waves
        // -- produce P chunk (each (row, j) computed exactly once in the block)
        for (int k = 0; k < NJ_CHUNK; k += 32) {
            const int j    = jc + k + lane;
            const float s  = psv + Edst[j];
            const float lr = s > 0.f ? s : ALPHA * s;
            const float v  = (parow[j] != 0) ? lr : NEG_INF;
            const float p  = __expf(v - pmx);     // unnormalized probability
            Pl[wave * P_STRIDE + k + lane] = p;
            dsum += p;
        }
        __syncthreads();
        // -- consume: acc(16x16) += P(16 x chunk) @ h(chunk x 16)
        for (int jt = 0; jt < NJ_CHUNK; jt += 16) {
#pragma unroll
            for (int kk = 0; kk < 16; kk += 4) {
                const int jl = jt + kk + koff;
                const v2f a = *(const v2f*)&Pl[col * P_STRIDE + jl];  // ds_load_b64
                v2f bb;
                const size_t gj = (size_t)(jc + jl) * OUT_F;
                bb.x = hb[gj];
                bb.y = hb[gj + OUT_F];
                acc = __builtin_amdgcn_wmma_f32_16x16x4_f32(false, a, false, bb,
                                                            (short)0, acc, false, false);
            }
        }
    }

    // ---- finish denominators (each lane holds a partial of row `wave`) ----
    for (int off = 16; off > 0; off >>= 1)
        dsum += __shfl_xor(dsum, off, 32);
    if (lane == 0) lds_d[wave] = dsum;
    __syncthreads();

    float* ob = out + (size_t)b * NNODES * OUT_F;
    for (int r = 0; r < 8; ++r) {
        const int rowC = r + hi * 8;
        const float val = fmaxf(acc[r] / lds_d[rowC], 0.f);
        ob[(size_t)(i0 + rowC) * OUT_F + n0 + col] = val;
    }
}

// ---------------------------------------------------------------------------
extern "C" void kernel_launch(void* const* d_in, const int* in_sizes, int n_in,
                              void* d_out, int out_size, void* d_ws, size_t ws_size,
                              hipStream_t stream) {
    (void)in_sizes; (void)n_in; (void)out_size; (void)ws_size;

    const float* inp  = (const float*)d_in[0];   // (8,2048,256) f32
    const int*   adj  = (const int*)  d_in[1];   // (2048,2048) i32
    const float* W    = (const float*)d_in[2];   // (256,256) f32
    const float* avec = (const float*)d_in[3];   // (512,1) f32
    float*       out  = (float*)d_out;           // (8,2048,256) f32

    float* h    = (float*)d_ws;                              // 16 MB
    float* esrc = h + (size_t)NBATCH * NNODES * OUT_F;       // 64 KB
    float* edst = esrc + (size_t)NBATCH * NNODES;            // 64 KB

    gat_h_kernel   <<<NBATCH * 128, 512, 0, stream>>>(inp, W, h);
    gat_e_kernel   <<<(NBATCH * NNODES) / 8, 256, 0, stream>>>(h, avec, esrc, edst);
    gat_attn_kernel<<<NBATCH * 128, 512, 0, stream>>>(h, adj, esrc, edst, out);
}